// C2CAttention_89154931131039
// MI455X (gfx1250) — compile-verified
//
#include <hip/hip_runtime.h>

typedef __attribute__((ext_vector_type(2))) float v2f;
typedef __attribute__((ext_vector_type(8))) float v8f;

constexpr int B = 32, C = 64, T = 30000, D = 64;
constexpr int TCHUNK = 16;
constexpr int NCHUNK = T / TCHUNK;                 // 1875 (exact)
constexpr int WPB = 8;                             // waves per block
constexpr int GROUPS = (NCHUNK + WPB - 1) / WPB;   // 235

// ---------------- kernel 1: desc[b][c] = mean_t x[b][c][t] ----------------
__global__ void desc_kernel(const float* __restrict__ x, float* __restrict__ desc) {
    __shared__ float red[256];
    const int bc = blockIdx.x;                       // b*C + c
    const float4* row = (const float4*)(x + (size_t)bc * T); // rows are 16B aligned (T*4 % 16 == 0)
    float s = 0.0f;
    for (int i = threadIdx.x; i < T / 4; i += 256) {
        float4 v = row[i];
        s += v.x + v.y + v.z + v.w;
    }
    red[threadIdx.x] = s;
    __syncthreads();
    for (int off = 128; off > 0; off >>= 1) {
        if ((int)threadIdx.x < off) red[threadIdx.x] += red[threadIdx.x + off];
        __syncthreads();
    }
    if (threadIdx.x == 0) desc[bc] = red[0] * (1.0f / T);
}

// ---------------- kernel 2: attn[b] = softmax(q k^T / 8) ----------------
__global__ void attn_kernel(const float* __restrict__ desc,
                            const float* __restrict__ Wq, const float* __restrict__ bq,
                            const float* __restrict__ Wk, const float* __restrict__ bk,
                            float* __restrict__ attn) {
    __shared__ float sdesc[C];
    const int b = blockIdx.x;
    const int c = threadIdx.x;                       // 0..63
    sdesc[c] = desc[b * C + c];
    __syncthreads();

    // logits[c][e] = sum_d (dc*Wq+bq)(de*Wk+bk) / 8 expands into 4 precomputed dots
    float A1 = 0.f, A2 = 0.f, A3 = 0.f, A4 = 0.f;
    for (int d = 0; d < D; ++d) {
        float wq = Wq[d], bqd = bq[d], wk = Wk[d], bkd = bk[d];
        A1 += wq * wk; A2 += wq * bkd; A3 += bqd * wk; A4 += bqd * bkd;
    }
    const float dc = sdesc[c];
    float l[C];
    float mx = -3.0e38f;
    for (int e = 0; e < C; ++e) {
        float de = sdesc[e];
        float v = (dc * de * A1 + dc * A2 + de * A3 + A4) * 0.125f;
        l[e] = v;
        mx = fmaxf(mx, v);
    }
    float s = 0.f;
    for (int e = 0; e < C; ++e) { float ev = __expf(l[e] - mx); l[e] = ev; s += ev; }
    const float inv = 1.0f / s;
    float* arow = attn + ((size_t)b * C + c) * C;
    for (int e = 0; e < C; ++e) arow[e] = l[e] * inv;
}

// ---------------- kernel 3: out = x + alpha * (attn @ x), WMMA FP32 ----------------
// One wave handles a 16-column chunk (all 64 output rows) of one batch.
__global__ void mix_kernel(const float* __restrict__ x, const float* __restrict__ attn,
                           const float* __restrict__ alpha_p, float* __restrict__ out) {
    __shared__ float sattn[C * C];                   // 16 KB: attn[b], shared by 8 waves
    const int b = blockIdx.x / GROUPS;
    const int g = blockIdx.x % GROUPS;

    for (int i = threadIdx.x; i < C * C; i += 256)
        sattn[i] = attn[(size_t)b * C * C + i];
    __syncthreads();

    const int wave = threadIdx.x >> 5;
    const int lane = threadIdx.x & 31;
    const int half = lane >> 4;                      // 0: lanes 0-15, 1: lanes 16-31
    const int lm   = lane & 15;
    const int chunk = g * WPB + wave;                // wave-uniform -> EXEC stays full
    if (chunk >= NCHUNK) return;

    const float alpha = *alpha_p;
    const size_t t0 = (size_t)chunk * TCHUNK;
    const float* xb = x + (size_t)b * C * T;
    float* ob = out + (size_t)b * C * T;

    v8f acc0 = {}, acc1 = {}, acc2 = {}, acc3 = {};
    // K=64 in 16 steps of 4. FP32 A 16x4 layout: lane holds A[lm][2*half + j] in VGPR j
    // => each lane's pair is contiguous in the attn row (single ds_load_b64).
    // FP32 B 4x16 layout (analogous): lane holds B[2*half + j][lm] in VGPR j.
    for (int ks = 0; ks < 16; ++ks) {
        const int kbase = ks * 4 + half * 2;
        v2f bf;
        bf.x = xb[(size_t)(kbase + 0) * T + t0 + lm];
        bf.y = xb[(size_t)(kbase + 1) * T + t0 + lm];
        v2f a0 = *(const v2f*)&sattn[( 0 + lm) * C + kbase];
        v2f a1 = *(const v2f*)&sattn[(16 + lm) * C + kbase];
        v2f a2 = *(const v2f*)&sattn[(32 + lm) * C + kbase];
        v2f a3 = *(const v2f*)&sattn[(48 + lm) * C + kbase];
        acc0 = __builtin_amdgcn_wmma_f32_16x16x4_f32(false, a0, false, bf, (short)0, acc0, false, false);
        acc1 = __builtin_amdgcn_wmma_f32_16x16x4_f32(false, a1, false, bf, (short)0, acc1, false, false);
        acc2 = __builtin_amdgcn_wmma_f32_16x16x4_f32(false, a2, false, bf, (short)0, acc2, false, false);
        acc3 = __builtin_amdgcn_wmma_f32_16x16x4_f32(false, a3, false, bf, (short)0, acc3, false, false);
    }

    // C/D layout: VGPR r, lane -> row (r + 8*half), col lm. Residual x reload hits L0
    // (same 4KB footprint as the B-fragment loads); out stores are nontemporal.
    v8f accs[4] = { acc0, acc1, acc2, acc3 };
#pragma unroll
    for (int mt = 0; mt < 4; ++mt) {
#pragma unroll
        for (int r = 0; r < 8; ++r) {
            const int m = 16 * mt + r + 8 * half;
            const size_t idx = (size_t)m * T + t0 + lm;
            const float v = xb[idx] + alpha * accs[mt][r];
            __builtin_nontemporal_store(v, &ob[idx]);
        }
    }
}

extern "C" void kernel_launch(void* const* d_in, const int* in_sizes, int n_in,
                              void* d_out, int out_size, void* d_ws, size_t ws_size,
                              hipStream_t stream) {
    const float* x     = (const float*)d_in[0];
    const float* Wq    = (const float*)d_in[1];
    const float* bq    = (const float*)d_in[2];
    const float* Wk    = (const float*)d_in[3];
    const float* bk    = (const float*)d_in[4];
    // d_in[5], d_in[6] = Wv, bv: computed but unused in the reference
    const float* alpha = (const float*)d_in[7];
    float* out = (float*)d_out;

    float* desc = (float*)d_ws;                 // B*C floats
    float* attn = desc + B * C;                 // B*C*C floats (~512 KB total ws use)

    desc_kernel<<<B * C, 256, 0, stream>>>(x, desc);
    attn_kernel<<<B, C, 0, stream>>>(desc, Wq, bq, Wk, bk, attn);
    mix_kernel<<<B * GROUPS, 256, 0, stream>>>(x, attn, alpha, out);
}